// GPUFeatureExtractor_78512002170896
// MI455X (gfx1250) — compile-verified
//
#include <hip/hip_runtime.h>
#include <math.h>

#define BB 32
#define HH 1024
#define WW 1024
#define WR 513
#define KPIX (HH * WR)          // 525312
#define NCHUNK (KPIX / 4)       // 131328
#define TWO_PI_F 6.2831853071795864769f

typedef float v2f __attribute__((ext_vector_type(2)));
typedef float v8f __attribute__((ext_vector_type(8)));

// async-to-LDS builtin pointer types (per hipcc diagnostic: param1 = AS1 int vector ptr)
typedef int vs4i __attribute__((vector_size(16)));
typedef int vs2i __attribute__((vector_size(8)));
typedef __attribute__((address_space(1))) vs4i g1v4i;
typedef __attribute__((address_space(3))) vs4i l3v4i;
typedef __attribute__((address_space(1))) vs2i g1v2i;
typedef __attribute__((address_space(3))) vs2i l3v2i;

#if __has_builtin(__builtin_amdgcn_global_load_async_to_lds_b128) && \
    __has_builtin(__builtin_amdgcn_global_load_async_to_lds_b64)
#define HAVE_ASYNC_LDS 1
#else
#define HAVE_ASYNC_LDS 0
#endif

__device__ inline void wait_async0() {
#if __has_builtin(__builtin_amdgcn_s_wait_asynccnt)
    __builtin_amdgcn_s_wait_asynccnt(0);
#else
    asm volatile("s_wait_asynccnt 0x0" ::: "memory");
#endif
}

// ---------------- Zernike radial-polynomial table (compile-time) ----------------
struct ZTab { float C[64][10]; float ms[64]; };

constexpr double zfact(int n) { double r = 1.0; for (int i = 2; i <= n; ++i) r *= (double)i; return r; }

constexpr ZTab makeZTab() {
    ZTab t = {};
    int idx = 0;
    for (int nn = 0; nn < 10; ++nn) {
        for (int m = -nn; m <= nn; m += 2) {
            int am = m < 0 ? -m : m;
            for (int k = 0; k <= (nn - am) / 2; ++k) {
                double c = ((k & 1) ? -1.0 : 1.0) * zfact(nn - k) /
                           (zfact(k) * zfact((nn + am) / 2 - k) * zfact((nn - am) / 2 - k));
                t.C[idx][nn - 2 * k] += (float)c;
            }
            t.ms[idx] = (float)m;
            ++idx;
        }
    }
    return t;
}

__constant__ ZTab g_ztab = makeZTab();

// ---------------- 1024-point in-LDS complex FFT (forward, e^{-i...}) ----------------
__device__ inline void fft1024(float2* s, int tid) {
    for (int i = tid; i < 1024; i += 256) {
        int j = (int)(__brev((unsigned)i) >> 22);   // reverse 10 bits
        if (j > i) { float2 tmp = s[i]; s[i] = s[j]; s[j] = tmp; }
    }
    __syncthreads();
    for (int len = 2; len <= 1024; len <<= 1) {
        int half = len >> 1;
        float ang = -TWO_PI_F / (float)len;
        for (int k = tid; k < 512; k += 256) {
            int grp = k / half;
            int j = k - grp * half;
            int i0 = grp * len + j;
            int i1 = i0 + half;
            float sw, cw;
            __sincosf(ang * (float)j, &sw, &cw);
            float2 u = s[i0], v = s[i1];
            float2 wv = make_float2(v.x * cw - v.y * sw, v.x * sw + v.y * cw);
            s[i0] = make_float2(u.x + wv.x, u.y + wv.y);
            s[i1] = make_float2(u.x - wv.x, u.y - wv.y);
        }
        __syncthreads();
    }
}

// ---------------- Pass 1: real FFT along W for every row ----------------
__global__ void k_rowfft(const float* __restrict__ x, float2* __restrict__ F) {
    __shared__ float2 s[1024];
    int row = blockIdx.x;                    // b*HH + h
    int tid = threadIdx.x;
    const float* xr = x + (size_t)row * WW;
#if HAVE_ASYNC_LDS
    __shared__ float tmp[1024];
    // async stage 16B per lane: 256 lanes x 16B = 4KB row, tracked by ASYNCcnt
    __builtin_amdgcn_global_load_async_to_lds_b128(
        (g1v4i*)(xr + tid * 4), (l3v4i*)&tmp[tid * 4], 0, 0);
    wait_async0();
    __syncthreads();
    for (int i = tid; i < 1024; i += 256) s[i] = make_float2(tmp[i], 0.0f);
#else
    for (int i = tid; i < 1024; i += 256) s[i] = make_float2(xr[i], 0.0f);
#endif
    __syncthreads();
    fft1024(s, tid);
    float2* out = F + (size_t)row * WR;
    for (int w = tid; w < WR; w += 256) out[w] = s[w];
}

// ---------------- Pass 2: complex FFT along H per column; emit phase + |.|max ----------------
__global__ void k_colfft(const float2* __restrict__ F, float* __restrict__ PU,
                         unsigned int* __restrict__ magmax) {
    __shared__ float2 s[1024];
    __shared__ float red[256];
    int col = blockIdx.x;                    // b*WR + w
    int b = col / WR, w = col - b * WR;
    int tid = threadIdx.x;
    const float2* base = F + (size_t)b * HH * WR + w;
#if HAVE_ASYNC_LDS
    #pragma unroll
    for (int h4 = 0; h4 < 4; ++h4) {
        int h = tid + h4 * 256;
        __builtin_amdgcn_global_load_async_to_lds_b64(
            (g1v2i*)(base + (size_t)h * WR), (l3v2i*)&s[h], 0, 0);
    }
    wait_async0();
#else
    for (int h = tid; h < 1024; h += 256) s[h] = base[(size_t)h * WR];
#endif
    __syncthreads();
    fft1024(s, tid);
    float mmax = 0.0f;
    float* pu = PU + (size_t)b * HH * WR + w;
    for (int h = tid; h < 1024; h += 256) {
        float2 c = s[h];
        float mag = sqrtf(c.x * c.x + c.y * c.y);
        mmax = fmaxf(mmax, mag);
        float ph = atan2f(c.y, c.x);
        if (ph < 0.0f) ph += TWO_PI_F;
        pu[(size_t)h * WR] = ph;
    }
    red[tid] = mmax;
    __syncthreads();
    for (int st = 128; st > 0; st >>= 1) {
        if (tid < st) red[tid] = fmaxf(red[tid], red[tid + st]);
        __syncthreads();
    }
    if (tid == 0) atomicMax(magmax + b, __float_as_uint(red[0]));  // mags >= 0: bit-monotonic
}

// ---------------- Accumulator init (deterministic per launch) ----------------
__global__ void k_init(float* coeffs, float* acc, unsigned* magmax, unsigned* maskcnt) {
    int i = blockIdx.x * 256 + threadIdx.x;
    if (i < BB * 64) coeffs[i] = 0.0f;
    if (i < BB * 4)  acc[i] = 0.0f;
    if (i < BB)      magmax[i] = 0u;
    if (i == 0)      *maskcnt = 0u;
}

// ---------------- Mask pixel count ----------------
__device__ inline float maskAt(int p) {
    int h = p / WR, w = p - h * WR;
    float xg = fmaf((float)w, 2.0f / 512.0f, -1.0f);
    float yg = fmaf((float)h, 2.0f / 1023.0f, -1.0f);
    return (xg * xg + yg * yg <= 1.0f) ? 1.0f : 0.0f;
}

__global__ void k_maskcnt(unsigned* maskcnt) {
    __shared__ unsigned red[256];
    int tid = threadIdx.x;
    int p = blockIdx.x * 256 + tid;
    unsigned m = 0;
    if (p < KPIX) m = (maskAt(p) > 0.5f) ? 1u : 0u;
    red[tid] = m;
    __syncthreads();
    for (int st = 128; st > 0; st >>= 1) {
        if (tid < st) red[tid] += red[tid + st];
        __syncthreads();
    }
    if (tid == 0 && red[0]) atomicAdd(maskcnt, red[0]);
}

// ---------------- Gradient statistics (central diff, one-sided edges) ----------------
__global__ void k_gradstats(const float* __restrict__ PU, float* __restrict__ acc) {
    __shared__ float red[256];
    int tid = threadIdx.x;
    int p = blockIdx.x * 256 + tid;
    int b = blockIdx.y;
    float gx = 0.0f, gy = 0.0f;
    if (p < KPIX) {
        int h = p / WR, w = p - h * WR;
        const float* pb = PU + (size_t)b * KPIX;
        if (w == 0)           gx = pb[h * WR + 1] - pb[h * WR];
        else if (w == WR - 1) gx = pb[h * WR + w] - pb[h * WR + w - 1];
        else                  gx = 0.5f * (pb[h * WR + w + 1] - pb[h * WR + w - 1]);
        if (h == 0)           gy = pb[WR + w] - pb[w];
        else if (h == HH - 1) gy = pb[h * WR + w] - pb[(h - 1) * WR + w];
        else                  gy = 0.5f * (pb[(h + 1) * WR + w] - pb[(h - 1) * WR + w]);
    }
    float vals[4] = { gx, gx * gx, gy, gy * gy };
    for (int q = 0; q < 4; ++q) {
        red[tid] = vals[q];
        __syncthreads();
        for (int st = 128; st > 0; st >>= 1) {
            if (tid < st) red[tid] += red[tid + st];
            __syncthreads();
        }
        if (tid == 0) atomicAdd(&acc[b * 4 + q], red[0]);
        __syncthreads();
    }
}

// ---------------- Zernike projection via V_WMMA_F32_16X16X4_F32 ----------------
// coeffs[b,t] = sum_p pm[b,p]*Z[t,p];  M = batch (2x16 tiles), N = 2 term-tiles per y-slice, K = pixels.
// Returns mask(p); fills z[j] = Z[(ntbase+j)*16+n, p]  (rows >= 55 have all-zero C -> z = 0)
__device__ inline float zpix(int p, int n, int ntbase, float z[2]) {
    int h = p / WR, w = p - h * WR;
    float xg = fmaf((float)w, 2.0f / 512.0f, -1.0f);
    float yg = fmaf((float)h, 2.0f / 1023.0f, -1.0f);
    float r2 = xg * xg + yg * yg;
    float rho = sqrtf(r2);
    float th = atan2f(yg, xg);
    float pw[10];
    pw[0] = 1.0f;
    #pragma unroll
    for (int i = 1; i < 10; ++i) pw[i] = pw[i - 1] * rho;
    #pragma unroll
    for (int j = 0; j < 2; ++j) {
        int t = (ntbase + j) * 16 + n;
        float R = 0.0f;
        #pragma unroll
        for (int q = 0; q < 10; ++q) R = fmaf(g_ztab.C[t][q], pw[q], R);
        float m = g_ztab.ms[t];
        z[j] = R * ((m >= 0.0f) ? cosf(m * th) : sinf(-m * th));
    }
    return (r2 <= 1.0f) ? 1.0f : 0.0f;
}

__global__ void k_zernike(const float* __restrict__ PU, float* __restrict__ coeffs) {
    int tid = threadIdx.x;
    int lane = tid & 31;
    int n = lane & 15;                       // M-row / N-col within a 16-tile
    int halfsel = lane >> 4;                 // K pair selector (0 -> K=0,1 ; 1 -> K=2,3)
    int waveId = blockIdx.x * (blockDim.x >> 5) + (tid >> 5);
    int nWaves = gridDim.x * (blockDim.x >> 5);
    int ntbase = blockIdx.y * 2;             // N-tile group: {0,1} or {2,3}

    v8f acc[2][2];                           // 32 accumulator VGPRs -> no spills
    #pragma unroll
    for (int mt = 0; mt < 2; ++mt)
        #pragma unroll
        for (int j = 0; j < 2; ++j)
            #pragma unroll
            for (int r = 0; r < 8; ++r) acc[mt][j][r] = 0.0f;

    for (int c = waveId; c < NCHUNK; c += nWaves) {   // wave-uniform loop, EXEC stays full
        int p0 = c * 4 + halfsel * 2;
        float z0[2], z1[2];
        float m0 = zpix(p0, n, ntbase, z0);
        float m1 = zpix(p0 + 1, n, ntbase, z1);

        v2f a[2], bf[2];
        #pragma unroll
        for (int mt = 0; mt < 2; ++mt) {
            const float* pb = PU + (size_t)(mt * 16 + n) * KPIX;
            float2 pv = *(const float2*)(pb + p0);    // p0 even -> 8B aligned
            a[mt].x = pv.x * m0;
            a[mt].y = pv.y * m1;
        }
        bf[0].x = z0[0]; bf[0].y = z1[0];
        bf[1].x = z0[1]; bf[1].y = z1[1];

        #pragma unroll
        for (int mt = 0; mt < 2; ++mt)
            #pragma unroll
            for (int j = 0; j < 2; ++j)
                acc[mt][j] = __builtin_amdgcn_wmma_f32_16x16x4_f32(
                    false, a[mt], false, bf[j], (short)0, acc[mt][j], false, false);
    }

    // C/D layout: VGPR r -> M=r (lanes 0-15), M=r+8 (lanes 16-31); N = lane%16
    #pragma unroll
    for (int mt = 0; mt < 2; ++mt)
        #pragma unroll
        for (int j = 0; j < 2; ++j)
            #pragma unroll
            for (int r = 0; r < 8; ++r) {
                int batch = mt * 16 + r + 8 * halfsel;
                int t = (ntbase + j) * 16 + n;
                atomicAdd(&coeffs[batch * 64 + t], acc[mt][j][r]);
            }
}

// ---------------- Finalize: [32,60] = 55 coeffs | gx.mean gy.mean gx.std gy.std | magmax ----------------
__global__ void k_final(const float* __restrict__ coeffs, const float* __restrict__ acc,
                        const unsigned* __restrict__ magmax, const unsigned* __restrict__ maskcnt,
                        float* __restrict__ out) {
    int i = blockIdx.x * 256 + threadIdx.x;
    if (i >= BB * 60) return;
    int b = i / 60, j = i - b * 60;
    const float N = (float)KPIX;
    float r;
    if (j < 55)      r = coeffs[b * 64 + j] / (float)(*maskcnt);
    else if (j == 55) r = acc[b * 4 + 0] / N;
    else if (j == 56) r = acc[b * 4 + 2] / N;
    else if (j == 57) {
        float s = acc[b * 4 + 0], s2 = acc[b * 4 + 1];
        r = sqrtf(fmaxf(0.0f, (s2 - s * s / N) / (N - 1.0f)));
    } else if (j == 58) {
        float s = acc[b * 4 + 2], s2 = acc[b * 4 + 3];
        r = sqrtf(fmaxf(0.0f, (s2 - s * s / N) / (N - 1.0f)));
    } else            r = __uint_as_float(magmax[b]);
    out[i] = r;
}

// ---------------- Launch ----------------
extern "C" void kernel_launch(void* const* d_in, const int* in_sizes, int n_in,
                              void* d_out, int out_size, void* d_ws, size_t ws_size,
                              hipStream_t stream) {
    (void)in_sizes; (void)n_in; (void)out_size; (void)ws_size;
    const float* x = (const float*)d_in[0];
    float* out = (float*)d_out;

    char* ws = (char*)d_ws;
    float2* F = (float2*)ws;                                   // 32*1024*513 cfloat = 134.5 MB
    size_t Fbytes = (size_t)BB * HH * WR * sizeof(float2);
    float* PU = (float*)(ws + Fbytes);                         // 32*1024*513 f32 = 67.2 MB (fits in L2)
    size_t PUbytes = (size_t)BB * HH * WR * sizeof(float);
    char* sm = ws + Fbytes + PUbytes;
    float* coeffs = (float*)sm;                                // 32*64 f32
    float* acc = coeffs + BB * 64;                             // 32*4 f32 (sgx, sgx2, sgy, sgy2)
    unsigned* magmax = (unsigned*)(acc + BB * 4);              // 32 u32 (float bits)
    unsigned* maskcnt = magmax + BB;                           // 1 u32

    k_init<<<8, 256, 0, stream>>>(coeffs, acc, magmax, maskcnt);
    k_rowfft<<<BB * HH, 256, 0, stream>>>(x, F);
    k_colfft<<<BB * WR, 256, 0, stream>>>(F, PU, magmax);
    k_maskcnt<<<(KPIX + 255) / 256, 256, 0, stream>>>(maskcnt);
    dim3 gs((KPIX + 255) / 256, BB);
    k_gradstats<<<gs, 256, 0, stream>>>(PU, acc);
    dim3 gz(256, 2);                                           // y: N-tile group, 2048 waves per slice
    k_zernike<<<gz, 256, 0, stream>>>(PU, coeffs);
    k_final<<<8, 256, 0, stream>>>(coeffs, acc, magmax, maskcnt, out);
}